// GNNModel_18038862644014
// MI455X (gfx1250) — compile-verified
//
#include <hip/hip_runtime.h>
#include <hip/hip_bf16.h>

// ---------------------------------------------------------------------------
// GAT (2 layers, H=4 heads, C=128) + MLP for MI455X (gfx1250, wave32).
// GEMMs: V_WMMA_F32_16X16X4_F32 (exact fp32 matrix pipe), 2x2 tiles/wave,
//        A panel staged to LDS via GLOBAL_LOAD_ASYNC_TO_LDS_B128 (ASYNCcnt).
// Edge softmax/scatter: hardware fp32 global atomics; h (41MB) is L2-resident
// (192MB L2) so the scatter phase is L2-bound.
// ---------------------------------------------------------------------------

typedef float v2f __attribute__((ext_vector_type(2)));
typedef float v8f __attribute__((ext_vector_type(8)));

#define HEADS 4
#define CH 128
#define DOUT 512   // HEADS*CH
#define NEG_SLOPE 0.2f

// 32-bit LDS byte address of a shared-memory pointer (for async-to-LDS ops).
__device__ __forceinline__ unsigned lds_byte_addr(const void* p) {
  return (unsigned)(unsigned long long)(__attribute__((address_space(3))) const void*)p;
}

// ------------------------- WMMA fp32 GEMM ----------------------------------
// Y[N,Dout] = X[N,K] @ W[K,Dout], row-major.  Block = 256 threads = 8 waves.
// Block output: 32 rows x 256 cols. Wave w owns a 32x32 region (2x2 WMMA
// tiles) -> 4 wmma per k-step sharing 2 A-frags + 2 B-frags.
// A panel (32 x K) lives in LDS with padded stride K+4 (conflict-free:
// lanes hit banks {4m+c}, all 32 distinct mod 64), filled by async DMA.
__global__ __launch_bounds__(256) void gemm_wmma_f32(
    const float* __restrict__ X, const float* __restrict__ W,
    float* __restrict__ Y, int K, int Dout) {
  extern __shared__ float Atile[];           // 32 * (K+4) floats
  const int LDSK = K + 4;                    // multiple of 4 -> 16B alignment
  const int tm = blockIdx.x;                 // 32-row tile index
  const int tid = threadIdx.x;

  // ---- async fill of the A panel: memory -> LDS, 16B per lane per issue ----
  {
    const int chunksPerRow = K >> 2;         // 4 floats per b128 chunk
    const int totalChunks = 32 * chunksPerRow;
    for (int ch = tid; ch < totalChunks; ch += 256) {
      int r = ch / chunksPerRow, c4 = (ch % chunksPerRow) << 2;
      unsigned long long gaddr =
          (unsigned long long)(X + (size_t)(tm * 32 + r) * K + c4);
      unsigned laddr = lds_byte_addr(&Atile[r * LDSK + c4]);
      asm volatile("global_load_async_to_lds_b128 %0, %1, off"
                   :: "v"(laddr), "v"(gaddr) : "memory");
    }
    asm volatile("s_wait_asynccnt 0" ::: "memory");
  }
  __syncthreads();

  const int wave = tid >> 5;
  const int lane = tid & 31;
  const int m  = lane & 15;                  // M (A) / N (B,C) within tile
  const int kh = lane >> 4;                  // K-half selector
  const int tnb = blockIdx.y * 256 + wave * 32;   // first output col of wave

  const float* wcol0 = W + (size_t)(tnb + m);
  const float* wcol1 = W + (size_t)(tnb + 16 + m);
  v8f a00 = {}, a01 = {}, a10 = {}, a11 = {};

  #pragma unroll 4
  for (int k = 0; k < K; k += 4) {
    const int kr = k + 2 * kh;
    // A frags: lane holds A[m][kr..kr+1] (row tiles 0 and 1)
    v2f af0 = *(const v2f*)&Atile[m * LDSK + kr];
    v2f af1 = *(const v2f*)&Atile[(16 + m) * LDSK + kr];
    // B frags: lane holds B[kr..kr+1][col] (col tiles 0 and 1)
    v2f bf0, bf1;
    bf0.x = wcol0[(size_t)(kr + 0) * Dout];
    bf0.y = wcol0[(size_t)(kr + 1) * Dout];
    bf1.x = wcol1[(size_t)(kr + 0) * Dout];
    bf1.y = wcol1[(size_t)(kr + 1) * Dout];
    a00 = __builtin_amdgcn_wmma_f32_16x16x4_f32(false, af0, false, bf0, (short)0, a00, false, false);
    a01 = __builtin_amdgcn_wmma_f32_16x16x4_f32(false, af0, false, bf1, (short)0, a01, false, false);
    a10 = __builtin_amdgcn_wmma_f32_16x16x4_f32(false, af1, false, bf0, (short)0, a10, false, false);
    a11 = __builtin_amdgcn_wmma_f32_16x16x4_f32(false, af1, false, bf1, (short)0, a11, false, false);
  }

  // C layout: VGPR r -> row = 8*kh + r (within 16-row tile), col = m
  const int c0 = tnb + m, c1 = tnb + 16 + m;
  #pragma unroll
  for (int r = 0; r < 8; ++r) {
    int row0 = tm * 32 + kh * 8 + r;         // row tile 0
    int row1 = row0 + 16;                    // row tile 1
    Y[(size_t)row0 * Dout + c0] = a00[r];
    Y[(size_t)row0 * Dout + c1] = a01[r];
    Y[(size_t)row1 * Dout + c0] = a10[r];
    Y[(size_t)row1 * Dout + c1] = a11[r];
  }
}

// ------------------------- attention scores --------------------------------
// a_s[n,h] = sum_c h[n,h,c]*att_src[h,c];  a_d likewise.
__global__ void att_scores_kernel(const float* __restrict__ h,
                                  const float* __restrict__ att_s,
                                  const float* __restrict__ att_d,
                                  float* __restrict__ a_s,
                                  float* __restrict__ a_d, int N) {
  int idx = blockIdx.x * blockDim.x + threadIdx.x;     // N*HEADS
  if (idx >= N * HEADS) return;
  int n = idx / HEADS, hh = idx % HEADS;
  const float* row = h + (size_t)n * DOUT + hh * CH;
  const float* as = att_s + hh * CH;
  const float* ad = att_d + hh * CH;
  float ss = 0.f, dd = 0.f;
  #pragma unroll 4
  for (int c = 0; c < CH; ++c) {
    float v = row[c];
    ss = fmaf(v, as[c], ss);
    dd = fmaf(v, ad[c], dd);
  }
  a_s[idx] = ss;
  a_d[idx] = dd;
}

// ---------------- order-preserving float<->uint for atomicMax --------------
__device__ __forceinline__ unsigned enc_f32(float f) {
  unsigned u = __float_as_uint(f);
  return (u & 0x80000000u) ? ~u : (u | 0x80000000u);
}
__device__ __forceinline__ float dec_f32(unsigned u) {
  unsigned b = (u & 0x80000000u) ? (u & 0x7FFFFFFFu) : ~u;
  return __uint_as_float(b);
}

__device__ __forceinline__ void edge_sd(const int* ei, int E, int e, int& s, int& d) {
  if (e < E) { s = ei[e]; d = ei[E + e]; }
  else       { s = d = e - E; }                // self-loops appended
}

// Pass A: segment max of leaky-relu logits (encoded-uint atomicMax).
__global__ void edge_max_kernel(const int* __restrict__ ei, int E, int N,
                                const float* __restrict__ a_s,
                                const float* __restrict__ a_d,
                                unsigned* __restrict__ mseg) {
  int ET = E + N;
  int t = blockIdx.x * blockDim.x + threadIdx.x;       // ET*HEADS
  if (t >= ET * HEADS) return;
  int e = t >> 2, hh = t & 3;
  int s, d; edge_sd(ei, E, e, s, d);
  float lg = a_s[s * HEADS + hh] + a_d[d * HEADS + hh];
  lg = (lg > 0.f) ? lg : NEG_SLOPE * lg;
  atomicMax(&mseg[d * HEADS + hh], enc_f32(lg));
}

// Pass B: e = exp(logit - max); store per-edge e; segment-sum via atomicAdd.
__global__ void edge_expsum_kernel(const int* __restrict__ ei, int E, int N,
                                   const float* __restrict__ a_s,
                                   const float* __restrict__ a_d,
                                   const unsigned* __restrict__ mseg,
                                   float* __restrict__ alpha,
                                   float* __restrict__ ssum) {
  int ET = E + N;
  int t = blockIdx.x * blockDim.x + threadIdx.x;
  if (t >= ET * HEADS) return;
  int e = t >> 2, hh = t & 3;
  int s, d; edge_sd(ei, E, e, s, d);
  float lg = a_s[s * HEADS + hh] + a_d[d * HEADS + hh];
  lg = (lg > 0.f) ? lg : NEG_SLOPE * lg;
  float ev = expf(lg - dec_f32(mseg[d * HEADS + hh]));
  alpha[(size_t)e * HEADS + hh] = ev;
  atomicAdd(&ssum[d * HEADS + hh], ev);
}

// Pass C: acc[dst] += h[src] * (alpha / (sum+eps)).  One block per edge,
// 256 threads cover the 512 channels (2 each) -> coalesced fp32 atomics.
__global__ __launch_bounds__(256) void edge_scatter_kernel(
    const int* __restrict__ ei, int E, int N,
    const float* __restrict__ h,
    const float* __restrict__ alpha,
    const float* __restrict__ ssum,
    float* __restrict__ acc) {
  int e = blockIdx.x;
  int s, d; edge_sd(ei, E, e, s, d);
  const float* hs = h + (size_t)s * DOUT;
  float* ad = acc + (size_t)d * DOUT;
  #pragma unroll
  for (int j = threadIdx.x; j < DOUT; j += 256) {
    int hh = j >> 7;                                   // j / CH
    float a = alpha[(size_t)e * HEADS + hh] /
              (ssum[d * HEADS + hh] + 1e-16f);
    atomicAdd(&ad[j], hs[j] * a);
  }
}

// ------------------------- epilogues ---------------------------------------
__device__ __forceinline__ float elu1(float v) {
  return v > 0.f ? v : (expf(v) - 1.f);
}

// layer1: acc = elu(acc + bias1)  (in place, [N,512])
__global__ void bias_elu_kernel(float* __restrict__ acc,
                                const float* __restrict__ bias, int total) {
  int idx = blockIdx.x * blockDim.x + threadIdx.x;
  if (idx >= total) return;
  acc[idx] = elu1(acc[idx] + bias[idx % DOUT]);
}

// layer2: z[n,c] = elu(mean_h acc[n,h,c] + bias2[c])   z is [N,128]
__global__ void mean_bias_elu_kernel(const float* __restrict__ acc,
                                     const float* __restrict__ bias,
                                     float* __restrict__ z, int N) {
  int idx = blockIdx.x * blockDim.x + threadIdx.x;     // N*CH
  if (idx >= N * CH) return;
  int n = idx / CH, c = idx % CH;
  const float* r = acc + (size_t)n * DOUT + c;
  float v = 0.25f * (r[0] + r[CH] + r[2 * CH] + r[3 * CH]) + bias[c];
  z[idx] = elu1(v);
}

// MLP: out[n] = relu(z[n]@fc1 + b1) @ fc2 + b2.  One wave32 per node,
// lane j owns hidden unit j (32 units), wave-reduce for the final dot.
__global__ __launch_bounds__(32) void mlp_kernel(
    const float* __restrict__ z,
    const float* __restrict__ fc1w, const float* __restrict__ fc1b,
    const float* __restrict__ fc2w, const float* __restrict__ fc2b,
    float* __restrict__ out) {
  int n = blockIdx.x;
  int j = threadIdx.x;                                  // 0..31
  const float* zr = z + (size_t)n * CH;
  float acc = fc1b[j];
  #pragma unroll 4
  for (int k = 0; k < CH; ++k)
    acc = fmaf(zr[k], fc1w[k * 32 + j], acc);           // fc1w is [128,32]
  acc = fmaxf(acc, 0.f);
  float v = acc * fc2w[j];
  #pragma unroll
  for (int off = 16; off > 0; off >>= 1)
    v += __shfl_xor(v, off, 32);
  if (j == 0) out[n] = v + fc2b[0];
}

// ---------------------------------------------------------------------------
extern "C" void kernel_launch(void* const* d_in, const int* in_sizes, int n_in,
                              void* d_out, int out_size, void* d_ws, size_t ws_size,
                              hipStream_t stream) {
  const float* x        = (const float*)d_in[0];
  const int*   ei       = (const int*)  d_in[1];   // [2,E] int32
  const float* W1       = (const float*)d_in[2];
  const float* att_src1 = (const float*)d_in[3];
  const float* att_dst1 = (const float*)d_in[4];
  const float* bias1    = (const float*)d_in[5];
  const float* W2       = (const float*)d_in[6];
  const float* att_src2 = (const float*)d_in[7];
  const float* att_dst2 = (const float*)d_in[8];
  const float* bias2    = (const float*)d_in[9];
  const float* fc1w     = (const float*)d_in[10];
  const float* fc1b     = (const float*)d_in[11];
  const float* fc2w     = (const float*)d_in[12];
  const float* fc2b     = (const float*)d_in[13];
  float* out = (float*)d_out;

  const int F_IN = 128;
  const int N  = in_sizes[0] / F_IN;                 // 20000
  const int E  = in_sizes[1] / 2;                    // 320000
  const int ET = E + N;                              // with self-loops

  // ---- workspace layout (256B aligned) ----
  char* wsb = (char*)d_ws;
  size_t off = 0;
  auto carve = [&](size_t bytes) -> char* {
    char* p = wsb + off;
    off = (off + bytes + 255) & ~(size_t)255;
    return p;
  };
  float*    h     = (float*)   carve((size_t)N * DOUT * 4);  // features / z
  float*    acc   = (float*)   carve((size_t)N * DOUT * 4);  // aggregation
  float*    a_s   = (float*)   carve((size_t)N * HEADS * 4);
  float*    a_d   = (float*)   carve((size_t)N * HEADS * 4);
  unsigned* mseg  = (unsigned*)carve((size_t)N * HEADS * 4);
  float*    ssum  = (float*)   carve((size_t)N * HEADS * 4);
  float*    alpha = (float*)   carve((size_t)ET * HEADS * 4);
  (void)ws_size; (void)n_in;

  const int nhBlocks = (N * HEADS + 255) / 256;
  const int ehBlocks = (ET * HEADS + 255) / 256;
  const dim3 gemmGrid(N / 32, DOUT / 256);           // (625, 2)

  // =================== Layer 1 ===================
  hipMemsetAsync(acc,  0, (size_t)N * DOUT * 4, stream);
  hipMemsetAsync(mseg, 0, (size_t)N * HEADS * 4, stream);
  hipMemsetAsync(ssum, 0, (size_t)N * HEADS * 4, stream);

  gemm_wmma_f32<<<gemmGrid, 256, 32 * (F_IN + 4) * 4, stream>>>(x, W1, h, F_IN, DOUT);
  att_scores_kernel<<<nhBlocks, 256, 0, stream>>>(h, att_src1, att_dst1, a_s, a_d, N);
  edge_max_kernel   <<<ehBlocks, 256, 0, stream>>>(ei, E, N, a_s, a_d, mseg);
  edge_expsum_kernel<<<ehBlocks, 256, 0, stream>>>(ei, E, N, a_s, a_d, mseg, alpha, ssum);
  edge_scatter_kernel<<<ET, 256, 0, stream>>>(ei, E, N, h, alpha, ssum, acc);
  bias_elu_kernel<<<(N * DOUT + 255) / 256, 256, 0, stream>>>(acc, bias1, N * DOUT);

  // =================== Layer 2 ===================
  // h2 = out1 @ W2 (out1 lives in acc); then acc is recycled as aggregator.
  gemm_wmma_f32<<<gemmGrid, 256, 32 * (DOUT + 4) * 4, stream>>>(acc, W2, h, DOUT, DOUT);
  hipMemsetAsync(acc,  0, (size_t)N * DOUT * 4, stream);
  hipMemsetAsync(mseg, 0, (size_t)N * HEADS * 4, stream);
  hipMemsetAsync(ssum, 0, (size_t)N * HEADS * 4, stream);

  att_scores_kernel<<<nhBlocks, 256, 0, stream>>>(h, att_src2, att_dst2, a_s, a_d, N);
  edge_max_kernel   <<<ehBlocks, 256, 0, stream>>>(ei, E, N, a_s, a_d, mseg);
  edge_expsum_kernel<<<ehBlocks, 256, 0, stream>>>(ei, E, N, a_s, a_d, mseg, alpha, ssum);
  edge_scatter_kernel<<<ET, 256, 0, stream>>>(ei, E, N, h, alpha, ssum, acc);

  // mean over heads + bias + elu -> z (reuse h buffer; h2 no longer needed)
  float* z = h;
  mean_bias_elu_kernel<<<(N * CH + 255) / 256, 256, 0, stream>>>(acc, bias2, z, N);

  // =================== MLP ===================
  mlp_kernel<<<N, 32, 0, stream>>>(z, fc1w, fc1b, fc2w, fc2b, out);
}